// STFormer_30459908063336
// MI455X (gfx1250) — compile-verified
//
#include <hip/hip_runtime.h>
#include <math.h>

#define N_NODES  207
#define T_STEPS  12
#define C_IN     4
#define BATCH    4
#define NT       (N_NODES * T_STEPS)   // 2484
#define D_MODEL  96
#define NHEAD    4
#define HD       24
#define FF_DIM   384
#define N_MASKT  1863
#define N_UNMASK 621
#define MAX_NEIGH 40
#define ATTN_SCALE 0.20412414523193154f   // 1/sqrt(24)
#define SQRT_D     9.797958971132712f     // sqrt(96)

typedef float v2f __attribute__((ext_vector_type(2)));
typedef float v8f __attribute__((ext_vector_type(8)));

// ---------------------------------------------------------------------------
// Adjacency: A = softmax(relu(nv1 @ nv2), axis=1); keep iff (A > 1/N) and
// strictly greater than the 41st-largest in its row (rank < MAX_NEIGH).
// ---------------------------------------------------------------------------
__global__ void adj_kernel(const float* __restrict__ nv1,
                           const float* __restrict__ nv2,
                           float* __restrict__ adpA) {
  __shared__ float a[N_NODES];
  __shared__ float stat[2];
  int i = blockIdx.x;
  int tid = threadIdx.x;
  for (int j = tid; j < N_NODES; j += blockDim.x) {
    float s = 0.f;
    for (int r = 0; r < 10; ++r) s += nv1[i * 10 + r] * nv2[r * N_NODES + j];
    a[j] = s > 0.f ? s : 0.f;
  }
  __syncthreads();
  if (tid == 0) {
    float mx = -1e30f;
    for (int j = 0; j < N_NODES; ++j) mx = fmaxf(mx, a[j]);
    float sum = 0.f;
    for (int j = 0; j < N_NODES; ++j) sum += __expf(a[j] - mx);
    stat[0] = mx; stat[1] = 1.f / sum;
  }
  __syncthreads();
  float mx = stat[0], inv = stat[1];
  for (int j = tid; j < N_NODES; j += blockDim.x) a[j] = __expf(a[j] - mx) * inv;
  __syncthreads();
  const float thr = 1.0f / (float)N_NODES;
  for (int j = tid; j < N_NODES; j += blockDim.x) {
    float aj = a[j];
    int cnt = 0;
    for (int l = 0; l < N_NODES; ++l) cnt += (a[l] > aj) ? 1 : 0;
    bool keep = (aj > thr) && (cnt < MAX_NEIGH);
    adpA[i * N_NODES + j] = keep ? aj : 0.f;
  }
}

// Encoder additive attention bias: 0 where connected, -1e30 where adpA == 0.
__global__ void encmask_kernel(const float* __restrict__ adpA,
                               const int* __restrict__ uidx,
                               float* __restrict__ Mb) {
  int idx = blockIdx.x * blockDim.x + threadIdx.x;
  if (idx >= N_UNMASK * N_UNMASK) return;
  int q = idx / N_UNMASK, k = idx % N_UNMASK;
  int nq = uidx[q] / T_STEPS, nk = uidx[k] / T_STEPS;
  Mb[idx] = (adpA[nq * N_NODES + nk] == 0.f) ? -1e30f : 0.f;
}

// token[b,nt,:] = x0*dim_w + dim_b + pos + tod*tod_w + tod_b + dow*dow_w + dow_b
__global__ void token_kernel(const float* __restrict__ hist,
                             const float* __restrict__ dim_w, const float* __restrict__ dim_b,
                             const float* __restrict__ pos,
                             const float* __restrict__ tod_w, const float* __restrict__ tod_b,
                             const float* __restrict__ dow_w, const float* __restrict__ dow_b,
                             float* __restrict__ token) {
  int row = blockIdx.x;            // b*NT + nt
  int d = threadIdx.x;             // 0..95
  int b = row / NT, nt = row % NT;
  int n = nt / T_STEPS, t = nt % T_STEPS;
  const float* hp = hist + (((size_t)b * T_STEPS + t) * N_NODES + n) * C_IN;
  float x0 = hp[0], dow = hp[2], tod = hp[3];
  token[(size_t)row * D_MODEL + d] =
      x0 * dim_w[d] + dim_b[d] + pos[(size_t)nt * D_MODEL + d] +
      tod * tod_w[d] + tod_b[d] + dow * dow_w[d] + dow_b[d];
}

// enc_in = token[:, unmasked_idx, :] * sqrt(D)
__global__ void gather_kernel(const float* __restrict__ token,
                              const int* __restrict__ uidx,
                              float* __restrict__ X) {
  int row = blockIdx.x;            // b*N_UNMASK + s
  int d = threadIdx.x;
  int b = row / N_UNMASK, s = row % N_UNMASK;
  X[(size_t)row * D_MODEL + d] =
      token[((size_t)b * NT + uidx[s]) * D_MODEL + d] * SQRT_D;
}

// ---------------------------------------------------------------------------
// fp32 GEMM via V_WMMA_F32_16X16X4_F32, register-blocked 32x32 per wave:
// 4 accumulators, each A-frag feeds 2 WMMAs and each B-frag feeds 2 WMMAs.
// Y[M,N] = X[M,K] @ W[K,N] + bias (+ReLU). K % 4 == 0, N % 32 == 0.
// M edge via row-clamped loads + guarded stores (EXEC stays full for WMMA).
// ---------------------------------------------------------------------------
__global__ void gemm_wmma_kernel(const float* __restrict__ X, const float* __restrict__ W,
                                 const float* __restrict__ bias, float* __restrict__ Y,
                                 int M, int K, int N, int relu) {
  int lane = threadIdx.x & 31;
  int l16 = lane & 15, half = lane >> 4;
  int m0 = blockIdx.x * 32, n0 = blockIdx.y * 32;
  int ra0 = m0 + l16;      if (ra0 >= M) ra0 = M - 1;
  int ra1 = m0 + 16 + l16; if (ra1 >= M) ra1 = M - 1;
  v8f acc00 = {}, acc01 = {}, acc10 = {}, acc11 = {};
  for (int k0 = 0; k0 < K; k0 += 4) {
    int kx = k0 + 2 * half;
    v2f a0, a1, b0, b1;
    a0.x = X[(size_t)ra0 * K + kx];
    a0.y = X[(size_t)ra0 * K + kx + 1];
    a1.x = X[(size_t)ra1 * K + kx];
    a1.y = X[(size_t)ra1 * K + kx + 1];
    b0.x = W[(size_t)kx * N + n0 + l16];
    b0.y = W[(size_t)(kx + 1) * N + n0 + l16];
    b1.x = W[(size_t)kx * N + n0 + 16 + l16];
    b1.y = W[(size_t)(kx + 1) * N + n0 + 16 + l16];
    acc00 = __builtin_amdgcn_wmma_f32_16x16x4_f32(false, a0, false, b0, (short)0, acc00, false, false);
    acc01 = __builtin_amdgcn_wmma_f32_16x16x4_f32(false, a0, false, b1, (short)0, acc01, false, false);
    acc10 = __builtin_amdgcn_wmma_f32_16x16x4_f32(false, a1, false, b0, (short)0, acc10, false, false);
    acc11 = __builtin_amdgcn_wmma_f32_16x16x4_f32(false, a1, false, b1, (short)0, acc11, false, false);
  }
  int nA = n0 + l16, nB = n0 + 16 + l16;
  float bv0 = bias ? bias[nA] : 0.f;
  float bv1 = bias ? bias[nB] : 0.f;
  for (int r = 0; r < 8; ++r) {
    int mA = m0 + r + 8 * half;
    if (mA < M) {
      float v0 = acc00[r] + bv0;
      float v1 = acc01[r] + bv1;
      if (relu) { v0 = fmaxf(v0, 0.f); v1 = fmaxf(v1, 0.f); }
      Y[(size_t)mA * N + nA] = v0;
      Y[(size_t)mA * N + nB] = v1;
    }
    int mB = m0 + 16 + r + 8 * half;
    if (mB < M) {
      float v0 = acc10[r] + bv0;
      float v1 = acc11[r] + bv1;
      if (relu) { v0 = fmaxf(v0, 0.f); v1 = fmaxf(v1, 0.f); }
      Y[(size_t)mB * N + nA] = v0;
      Y[(size_t)mB * N + nB] = v1;
    }
  }
}

// ---------------------------------------------------------------------------
// Fused attention per (b, head, 16-row q-tile).
// Scores for the whole key range are staged in LDS (16 x Spad), softmax'd
// in-place (all 32 lanes: lane = (row, key-parity), halves merged with a
// wave32 shfl_xor), then P@V runs through WMMA.
// Q/K/V layout: [(b*S+s)*96 + h*24 + d].
// ---------------------------------------------------------------------------
__global__ void attn_kernel(const float* __restrict__ Q, const float* __restrict__ Kb,
                            const float* __restrict__ Vb, float* __restrict__ O,
                            const float* __restrict__ maskb, int S, int Spad) {
  extern __shared__ float sc[];    // [16][Spad]
  int lane = threadIdx.x & 31;
  int l16 = lane & 15, half = lane >> 4;
  int qt = blockIdx.x, h = blockIdx.y, b = blockIdx.z;
  int qrow = qt * 16 + l16; int qrc = qrow < S ? qrow : S - 1;
  const size_t hb = (size_t)h * HD;

  // Q fragments (A-matrix 16x24): lane row = l16, x/y = K sub-indices
  v2f aq[6];
  {
    const float* qp = Q + ((size_t)b * S + qrc) * D_MODEL + hb;
    for (int j = 0; j < 6; ++j) {
      int kd = 4 * j + 2 * half;
      aq[j].x = qp[kd];
      aq[j].y = qp[kd + 1];
    }
  }

  int ktiles = Spad / 16;
  // -------- S = Q K^T, tile by tile into LDS --------
  for (int kt = 0; kt < ktiles; ++kt) {
    int key = kt * 16 + l16; if (key >= S) key = S - 1;
    const float* kp = Kb + ((size_t)b * S + key) * D_MODEL + hb;
    v8f acc = {};
    for (int j = 0; j < 6; ++j) {
      int kd = 4 * j + 2 * half;
      v2f bf; bf.x = kp[kd]; bf.y = kp[kd + 1];
      acc = __builtin_amdgcn_wmma_f32_16x16x4_f32(false, aq[j], false, bf, (short)0,
                                                  acc, false, false);
    }
    for (int r = 0; r < 8; ++r)
      sc[(size_t)(r + 8 * half) * Spad + kt * 16 + l16] = acc[r];
  }
  __syncthreads();

  // -------- softmax rows: lane (l16, half); half handles keys of its parity
  {
    int m = l16;
    float* row = sc + (size_t)m * Spad;
    int qg = qt * 16 + m; if (qg >= S) qg = S - 1;
    const float* mrow = maskb ? (maskb + (size_t)qg * S) : (const float*)0;
    float mx = -1e30f;
    for (int k = half; k < S; k += 2) {
      float v = row[k] * ATTN_SCALE + (mrow ? mrow[k] : 0.f);
      row[k] = v;
      mx = fmaxf(mx, v);
    }
    mx = fmaxf(mx, __shfl_xor(mx, 16, 32));
    float sum = 0.f;
    for (int k = half; k < S; k += 2) {
      float e = __expf(row[k] - mx);
      row[k] = e;
      sum += e;
    }
    sum += __shfl_xor(sum, 16, 32);
    float inv = 1.f / sum;
    for (int k = half; k < S; k += 2) row[k] *= inv;
    for (int k = S + half; k < Spad; k += 2) row[k] = 0.f;  // kill pad for P@V
  }
  __syncthreads();

  // -------- O = P V (HD=24 -> one full 16-wide tile + one 8-wide tile) ----
  v8f o0 = {}, o1 = {};
  int hd2 = 16 + l16; if (hd2 >= HD) hd2 = HD - 1;  // clamp; masked on store
  for (int kt = 0; kt < ktiles; ++kt) {
    for (int ks = 0; ks < 4; ++ks) {
      int k0 = kt * 16 + ks * 4;
      v2f ap;
      ap.x = sc[(size_t)l16 * Spad + k0 + 2 * half];
      ap.y = sc[(size_t)l16 * Spad + k0 + 2 * half + 1];
      int keyx = k0 + 2 * half;     if (keyx >= S) keyx = S - 1;
      int keyy = k0 + 2 * half + 1; if (keyy >= S) keyy = S - 1;
      const float* vpx = Vb + ((size_t)b * S + keyx) * D_MODEL + hb;
      const float* vpy = Vb + ((size_t)b * S + keyy) * D_MODEL + hb;
      v2f b0, b1;
      b0.x = vpx[l16]; b0.y = vpy[l16];
      b1.x = vpx[hd2]; b1.y = vpy[hd2];
      o0 = __builtin_amdgcn_wmma_f32_16x16x4_f32(false, ap, false, b0, (short)0, o0,
                                                 false, false);
      o1 = __builtin_amdgcn_wmma_f32_16x16x4_f32(false, ap, false, b1, (short)0, o1,
                                                 false, false);
    }
  }
  for (int r = 0; r < 8; ++r) {
    int m = qt * 16 + r + 8 * half;
    if (m < S) {
      float* op = O + ((size_t)b * S + m) * D_MODEL + hb;
      op[l16] = o0[r];
      if (l16 < 8) op[16 + l16] = o1[r];
    }
  }
}

// LayerNorm over D=96, optional fused residual R; in-place on X (row-local).
__global__ void ln_kernel(float* __restrict__ X, const float* __restrict__ R,
                          const float* __restrict__ g, const float* __restrict__ bta) {
  __shared__ float red[D_MODEL];
  __shared__ float stat[2];
  int row = blockIdx.x;
  int d = threadIdx.x;
  size_t idx = (size_t)row * D_MODEL + d;
  float v = X[idx] + (R ? R[idx] : 0.f);
  red[d] = v;
  __syncthreads();
  if (d == 0) {
    float s = 0.f;
    for (int j = 0; j < D_MODEL; ++j) s += red[j];
    stat[0] = s / (float)D_MODEL;
  }
  __syncthreads();
  float m = stat[0];
  float c = v - m;
  red[d] = c * c;
  __syncthreads();
  if (d == 0) {
    float s = 0.f;
    for (int j = 0; j < D_MODEL; ++j) s += red[j];
    stat[1] = rsqrtf(s / (float)D_MODEL + 1e-5f);
  }
  __syncthreads();
  X[idx] = c * stat[1] * g[d] + bta[d];
}

// Decoder input: first 621 rows = e2d(enc) * sqrt(D)
__global__ void dec_assemble_enc(const float* __restrict__ E, float* __restrict__ Xd) {
  int row = blockIdx.x;            // b*N_UNMASK + s
  int d = threadIdx.x;
  int b = row / N_UNMASK, s = row % N_UNMASK;
  Xd[((size_t)b * NT + s) * D_MODEL + d] = E[(size_t)row * D_MODEL + d] * SQRT_D;
}

// Decoder input: masked rows = (mask_token + pos + tod/dow emb) * sqrt(D)
__global__ void dec_assemble_mask(const float* __restrict__ hist,
                                  const int* __restrict__ midx,
                                  const float* __restrict__ mask_token,
                                  const float* __restrict__ pos,
                                  const float* __restrict__ tod_w, const float* __restrict__ tod_b,
                                  const float* __restrict__ dow_w, const float* __restrict__ dow_b,
                                  float* __restrict__ Xd) {
  int row = blockIdx.x;            // b*N_MASKT + j
  int d = threadIdx.x;
  int b = row / N_MASKT, j = row % N_MASKT;
  int nt = midx[j]; int n = nt / T_STEPS, t = nt % T_STEPS;
  const float* hp = hist + (((size_t)b * T_STEPS + t) * N_NODES + n) * C_IN;
  float dow = hp[2], tod = hp[3];
  float v = mask_token[d] + pos[(size_t)nt * D_MODEL + d] +
            tod * tod_w[d] + tod_b[d] + dow * dow_w[d] + dow_b[d];
  Xd[((size_t)b * NT + N_UNMASK + j) * D_MODEL + d] = v * SQRT_D;
}

// recon_masked = (decoder rows >= N_UNMASK) @ out_w + out_b ; label gather.
__global__ void out_kernel(const float* __restrict__ Xd, const float* __restrict__ hist,
                           const int* __restrict__ midx,
                           const float* __restrict__ out_w, const float* __restrict__ out_b,
                           float* __restrict__ out) {
  __shared__ float red[D_MODEL];
  int row = blockIdx.x;            // b*N_MASKT + j
  int d = threadIdx.x;
  int b = row / N_MASKT, j = row % N_MASKT;
  size_t xr = ((size_t)b * NT + N_UNMASK + j) * D_MODEL;
  red[d] = Xd[xr + d] * out_w[d];
  __syncthreads();
  if (d == 0) {
    float s = 0.f;
    for (int i = 0; i < D_MODEL; ++i) s += red[i];
    out[row] = s + out_b[0];
    int nt = midx[j]; int n = nt / T_STEPS, t = nt % T_STEPS;
    out[BATCH * N_MASKT + row] =
        hist[(((size_t)b * T_STEPS + t) * N_NODES + n) * C_IN + 0];
  }
}

// ---------------------------------------------------------------------------
extern "C" void kernel_launch(void* const* d_in, const int* in_sizes, int n_in,
                              void* d_out, int out_size, void* d_ws, size_t ws_size,
                              hipStream_t stream) {
  (void)in_sizes; (void)n_in; (void)out_size; (void)ws_size;
  const float* hist  = (const float*)d_in[0];
  const int*   uidx  = (const int*)d_in[1];
  const int*   midx  = (const int*)d_in[2];
  const float* nv1   = (const float*)d_in[3];
  const float* nv2   = (const float*)d_in[4];
  const float* dim_w = (const float*)d_in[5];
  const float* dim_b = (const float*)d_in[6];
  const float* pos   = (const float*)d_in[7];
  const float* tod_w = (const float*)d_in[8];
  const float* tod_b = (const float*)d_in[9];
  const float* dow_w = (const float*)d_in[10];
  const float* dow_b = (const float*)d_in[11];
  const float* mtok  = (const float*)d_in[12];
  const float* eaw   = (const float*)d_in[13];
  const float* eab   = (const float*)d_in[14];
  const float* ew1   = (const float*)d_in[15];
  const float* eb1   = (const float*)d_in[16];
  const float* ew2   = (const float*)d_in[17];
  const float* eb2   = (const float*)d_in[18];
  const float* eln   = (const float*)d_in[19];
  const float* eng   = (const float*)d_in[20];
  const float* enb   = (const float*)d_in[21];
  const float* e2dw  = (const float*)d_in[22];
  const float* e2db  = (const float*)d_in[23];
  const float* daw   = (const float*)d_in[24];
  const float* dab   = (const float*)d_in[25];
  const float* dw1   = (const float*)d_in[26];
  const float* db1   = (const float*)d_in[27];
  const float* dw2   = (const float*)d_in[28];
  const float* db2   = (const float*)d_in[29];
  const float* dln   = (const float*)d_in[30];
  const float* dng   = (const float*)d_in[31];
  const float* dnb   = (const float*)d_in[32];
  const float* outw  = (const float*)d_in[33];
  const float* outb  = (const float*)d_in[34];
  float* out = (float*)d_out;

  float* ws = (float*)d_ws;
  size_t off = 0;
  auto alloc = [&](size_t n) { float* p = ws + off; off += n; return p; };
  const size_t MD = (size_t)BATCH * NT;   // max rows (decoder): 9936
  float* adpA  = alloc((size_t)N_NODES * N_NODES);
  float* Menc  = alloc((size_t)N_UNMASK * N_UNMASK);
  float* token = alloc((size_t)BATCH * NT * D_MODEL);
  float* X  = alloc(MD * D_MODEL);
  float* Qb = alloc(MD * D_MODEL);
  float* Kb = alloc(MD * D_MODEL);
  float* Vb = alloc(MD * D_MODEL);
  float* Ob = alloc(MD * D_MODEL);
  float* T1 = alloc(MD * D_MODEL);
  float* F1 = alloc(MD * FF_DIM);
  float* Xd = alloc(MD * D_MODEL);

  adj_kernel<<<N_NODES, 256, 0, stream>>>(nv1, nv2, adpA);
  {
    int tot = N_UNMASK * N_UNMASK;
    encmask_kernel<<<(tot + 255) / 256, 256, 0, stream>>>(adpA, uidx, Menc);
  }
  token_kernel<<<BATCH * NT, D_MODEL, 0, stream>>>(hist, dim_w, dim_b, pos, tod_w,
                                                   tod_b, dow_w, dow_b, token);
  gather_kernel<<<BATCH * N_UNMASK, D_MODEL, 0, stream>>>(token, uidx, X);

  auto gemm = [&](const float* A, const float* W, const float* bias, float* Y,
                  int M, int K, int N, int relu) {
    dim3 g((unsigned)((M + 31) / 32), (unsigned)(N / 32));
    gemm_wmma_kernel<<<g, 32, 0, stream>>>(A, W, bias, Y, M, K, N, relu);
  };

  auto layer = [&](float* H, int S, const float* aw, const float* ab,
                   const float* w1, const float* b1, const float* w2, const float* b2,
                   const float* ln, const float* maskb) {
    int M = BATCH * S;
    int Spad = ((S + 15) / 16) * 16;
    gemm(H, aw + 0 * D_MODEL * D_MODEL, ab + 0 * D_MODEL, Qb, M, D_MODEL, D_MODEL, 0);
    gemm(H, aw + 1 * D_MODEL * D_MODEL, ab + 1 * D_MODEL, Kb, M, D_MODEL, D_MODEL, 0);
    gemm(H, aw + 2 * D_MODEL * D_MODEL, ab + 2 * D_MODEL, Vb, M, D_MODEL, D_MODEL, 0);
    dim3 ag((unsigned)(Spad / 16), NHEAD, BATCH);
    size_t sh = (size_t)16 * Spad * sizeof(float);
    attn_kernel<<<ag, 32, sh, stream>>>(Qb, Kb, Vb, Ob, maskb, S, Spad);
    gemm(Ob, aw + 3 * D_MODEL * D_MODEL, ab + 3 * D_MODEL, T1, M, D_MODEL, D_MODEL, 0);
    ln_kernel<<<M, D_MODEL, 0, stream>>>(H, T1, ln + 0 * 2 * D_MODEL,
                                         ln + 0 * 2 * D_MODEL + D_MODEL);
    gemm(H, w1, b1, F1, M, D_MODEL, FF_DIM, 1);
    gemm(F1, w2, b2, T1, M, FF_DIM, D_MODEL, 0);
    ln_kernel<<<M, D_MODEL, 0, stream>>>(H, T1, ln + 1 * 2 * D_MODEL,
                                         ln + 1 * 2 * D_MODEL + D_MODEL);
  };

  // ---- encoder (4 layers, masked attention over 621 tokens) ----
  for (int l = 0; l < 4; ++l) {
    layer(X, N_UNMASK,
          eaw + (size_t)l * 4 * D_MODEL * D_MODEL, eab + (size_t)l * 4 * D_MODEL,
          ew1 + (size_t)l * D_MODEL * FF_DIM,      eb1 + (size_t)l * FF_DIM,
          ew2 + (size_t)l * FF_DIM * D_MODEL,      eb2 + (size_t)l * D_MODEL,
          eln + (size_t)l * 2 * 2 * D_MODEL, Menc);
  }
  ln_kernel<<<BATCH * N_UNMASK, D_MODEL, 0, stream>>>(X, nullptr, eng, enb);
  gemm(X, e2dw, e2db, T1, BATCH * N_UNMASK, D_MODEL, D_MODEL, 0);
  dec_assemble_enc<<<BATCH * N_UNMASK, D_MODEL, 0, stream>>>(T1, Xd);
  dec_assemble_mask<<<BATCH * N_MASKT, D_MODEL, 0, stream>>>(hist, midx, mtok, pos,
                                                             tod_w, tod_b, dow_w, dow_b, Xd);
  // ---- decoder (1 layer, full attention over 2484 tokens) ----
  layer(Xd, NT, daw, dab, dw1, db1, dw2, db2, dln, nullptr);
  ln_kernel<<<BATCH * NT, D_MODEL, 0, stream>>>(Xd, nullptr, dng, dnb);
  out_kernel<<<BATCH * N_MASKT, D_MODEL, 0, stream>>>(Xd, hist, midx, outw, outb, out);
}